// ROIPoolingLayer_23433341567204
// MI455X (gfx1250) — compile-verified
//
#include <hip/hip_runtime.h>

// ROI max-pooling, exact replica of the JAX reference binning.
// B=256, H=W=48, C=256, PH=PW=7. Purely HBM-bandwidth bound (~0.25 GB read
// once + 12.8 MB written; ~11 us at 23.3 TB/s). Strategy: one 128-thread
// block (4 wave32) per (b, ph, pw) bin. 32 lanes each own two channel quads
// (lane*4 and 128+lane*4) -> two fully-coalesced 512 B B128 NT loads per
// pixel; x-loop unrolled by 2 with dual accumulators -> 4 independent loads
// in flight per thread. 4 row-groups stripe the bin's rows; LDS b128 4-way
// max reduce; NT stores.

#define BB  256
#define HH  48
#define WW  48
#define CC  256
#define PHB 7
#define PWB 7

typedef __attribute__((ext_vector_type(4))) float v4f;

__device__ __forceinline__ v4f vmax4(v4f a, v4f b) {
    v4f r;
    r.x = fmaxf(a.x, b.x);
    r.y = fmaxf(a.y, b.y);
    r.z = fmaxf(a.z, b.z);
    r.w = fmaxf(a.w, b.w);
    return r;
}

__device__ __forceinline__ v4f vneg_inf() {
    v4f r;
    r.x = -INFINITY; r.y = -INFINITY; r.z = -INFINITY; r.w = -INFINITY;
    return r;
}

// ends[j] of the reference _bin_ends; j == -1 -> 0 (implicit start).
// w = s // nbins, k = nbins - w,
// ends[j] = (j+1)*w if j < k else k*w + (j-k+1)*(w+1)
__device__ __forceinline__ int bin_end(int j, int w, int k) {
    if (j < 0) return 0;
    return (j < k) ? (j + 1) * w : k * w + (j - k + 1) * (w + 1);
}

__global__ __launch_bounds__(128) void roipool_max_kernel(
    const float* __restrict__ fm,    // (B, H, W, C) f32
    const float* __restrict__ rois,  // (B, 4) f32
    float* __restrict__ out)         // (B, PH, PW, C) f32
{
    const int pw = (int)blockIdx.x % PWB;
    const int ph = (int)blockIdx.x / PWB;
    const int b  = (int)blockIdx.y;

    const int lane = (int)threadIdx.x & 31;  // quad A: ch 4*lane.., quad B: ch 128+4*lane..
    const int g    = (int)threadIdx.x >> 5;  // row group 0..3

    // ROI corners, truncation matches .astype(int32) on non-negative values.
    const float r0 = rois[b * 4 + 0];
    const float r1 = rois[b * 4 + 1];
    const float r2 = rois[b * 4 + 2];
    const float r3 = rois[b * 4 + 3];
    const int x0 = (int)((float)WW * r0);
    const int y0 = (int)((float)HH * r1);
    const int x1 = (int)((float)WW * r2);
    const int y1 = (int)((float)HH * r3);

    const int sx = x1 - x0;
    const int sy = y1 - y0;
    const int wx = sx / PWB, kx = PWB - wx;
    const int wy = sy / PHB, ky = PHB - wy;

    // This block's bin in ROI-relative coords, clipped to the valid region
    // (pixels with r >= s map to the dump bin in the reference -> excluded).
    int xlo = bin_end(pw - 1, wx, kx);
    int xhi = bin_end(pw,     wx, kx); if (xhi > sx) xhi = sx;
    int ylo = bin_end(ph - 1, wy, ky);
    int yhi = bin_end(ph,     wy, ky); if (yhi > sy) yhi = sy;

    // segment_max identity for (theoretically) empty bins is -inf.
    v4f accA0 = vneg_inf(), accB0 = vneg_inf();   // even x pixels
    v4f accA1 = vneg_inf(), accB1 = vneg_inf();   // odd  x pixels

    const size_t cA = (size_t)lane * 4;       // quad A channel offset
    const size_t cB = cA + 128;               // quad B channel offset
    const size_t base_b = (size_t)b * (HH * WW * CC);

    for (int ry = ylo + g; ry < yhi; ry += 4) {
        const float* row = fm + base_b + (size_t)(y0 + ry) * (WW * CC);
        int rx = xlo;
        for (; rx + 1 < xhi; rx += 2) {
            const float* p0 = row + (size_t)(x0 + rx) * CC;
            const float* p1 = p0 + CC;
            // 4 independent B128 NT loads issued before any use.
            const v4f vA0 = __builtin_nontemporal_load((const v4f*)(p0 + cA));
            const v4f vB0 = __builtin_nontemporal_load((const v4f*)(p0 + cB));
            const v4f vA1 = __builtin_nontemporal_load((const v4f*)(p1 + cA));
            const v4f vB1 = __builtin_nontemporal_load((const v4f*)(p1 + cB));
            accA0 = vmax4(accA0, vA0);
            accB0 = vmax4(accB0, vB0);
            accA1 = vmax4(accA1, vA1);
            accB1 = vmax4(accB1, vB1);
        }
        if (rx < xhi) {
            const float* p0 = row + (size_t)(x0 + rx) * CC;
            const v4f vA0 = __builtin_nontemporal_load((const v4f*)(p0 + cA));
            const v4f vB0 = __builtin_nontemporal_load((const v4f*)(p0 + cB));
            accA0 = vmax4(accA0, vA0);
            accB0 = vmax4(accB0, vB0);
        }
    }

    v4f accA = vmax4(accA0, accA1);
    v4f accB = vmax4(accB0, accB1);

    // 4-way cross-wave-group max reduce through LDS (b128 traffic).
    // Layout: [0..127] = quad A per thread, [128..255] = quad B per thread.
    __shared__ v4f smem[256];
    smem[threadIdx.x]       = accA;
    smem[threadIdx.x + 128] = accB;
    __syncthreads();

    if (threadIdx.x < 32) {
        v4f a = vmax4(vmax4(smem[lane],       smem[lane + 32]),
                      vmax4(smem[lane + 64],  smem[lane + 96]));
        v4f bq = vmax4(vmax4(smem[lane + 128], smem[lane + 160]),
                       vmax4(smem[lane + 192], smem[lane + 224]));

        float* o = out + (((size_t)b * PHB + ph) * PWB + pw) * CC;
        __builtin_nontemporal_store(a,  (v4f*)(o + cA));
        __builtin_nontemporal_store(bq, (v4f*)(o + cB));
    }
}

extern "C" void kernel_launch(void* const* d_in, const int* in_sizes, int n_in,
                              void* d_out, int out_size, void* d_ws, size_t ws_size,
                              hipStream_t stream) {
    const float* fm   = (const float*)d_in[0];  // (256, 48, 48, 256) f32
    const float* rois = (const float*)d_in[1];  // (256, 4) f32
    float* out = (float*)d_out;                 // (256, 7, 7, 256) f32

    dim3 grid(PHB * PWB, BB);   // one block per (b, ph, pw) bin
    dim3 block(128);            // 4 wave32: 32 lanes x 4 row groups
    roipool_max_kernel<<<grid, block, 0, stream>>>(fm, rois, out);
}